// TVLeakyReLU_22823456211566
// MI455X (gfx1250) — compile-verified
//
#include <hip/hip_runtime.h>
#include <hip/hip_bf16.h>
#include <stdint.h>

// ---------------------------------------------------------------------------
// TV / leaky-ReLU denoiser, LDS-resident Chambolle iterations for gfx1250.
//
//  * one workgroup = one 64x64 output tile of one 256x256 plane
//  * 84x84 region (halo = 10 = NB_ITERATIONS) of z async-DMA'd into LDS
//    (GLOBAL_LOAD_ASYNC_TO_LDS_B32, ASYNCcnt)
//  * all 10 iterations run in LDS: u, px, py, z resident
//    (4 * 84*84 * 4B = 110.25 KB < 320 KB/WGP  -> 2 WGs per WGP)
//  * result tile written back with ONE Tensor Data Mover descriptor per
//    workgroup (TENSOR_STORE_FROM_LDS, TENSORcnt)
// ---------------------------------------------------------------------------

#define ALPHA_  0.2f
#define TAU_    0.25f
#define EPS_    1e-12f
#define NITER   10
#define TILE    64
#define HALO    10                 // == NITER: valid region shrinks 1/side/iter
#define R       (TILE + 2*HALO)    // 84
#define RR      (R*R)              // 7056 elements per LDS array
#define PW      256
#define PH      256
#define PLANE   (PW*PH)
#define NTHR    256
#define SMEM_BYTES (4u * RR * sizeof(float))   // u, px, py, z = 112896 B

typedef uint32_t u32x4 __attribute__((ext_vector_type(4)));
typedef uint32_t u32x8 __attribute__((ext_vector_type(8)));

// ---- CDNA5 async LDS <-> global (ASYNCcnt) --------------------------------
__device__ __forceinline__ uint32_t lds_byte_off(const void* p) {
    // generic LDS pointer: low 32 bits are the LDS byte address (ISA 10.2)
    return (uint32_t)(uintptr_t)p;
}
__device__ __forceinline__ void async_load_b32(uint32_t lds_addr,
                                               const float* sbase,
                                               uint32_t voff) {
    // GVS mode: mem_addr = SGPR64 + VGPR32, LDS dest address in VGPR %0
    asm volatile("global_load_async_to_lds_b32 %0, %1, %2"
                 :: "v"(lds_addr), "v"(voff), "s"(sbase) : "memory");
}
__device__ __forceinline__ void wait_async0() {
    asm volatile("s_wait_asynccnt 0" ::: "memory");
}

// ---- CDNA5 Tensor Data Mover: one-shot 2D tile store (TENSORcnt) ----------
// D# group0 (4 SGPRs): [1:0]=count=1, [63:32]=lds_addr,
//                      [120:64]=global tile addr, [127:126]=type=2
// D# group1 (8 SGPRs): [17:16]=data_size(2 -> 4B), [79:48]=tensor_dim0,
//                      [111:80]=tensor_dim1, [127:112]=tile_dim0,
//                      [143:128]=tile_dim1, [207:160]=tensor_dim0_stride,
//                      workgroup_mask/pad/iterate = 0 (not in a cluster)
__device__ __forceinline__ void tdm_store_tile64(uint32_t lds_addr,
                                                 float* gtile) {
    const uint64_t ga = (uint64_t)(uintptr_t)gtile;
    u32x4 g0;
    g0.x = 1u;                                    // count = 1 valid descriptor
    g0.y = lds_addr;                              // contiguous LDS source
    g0.z = (uint32_t)ga;                          // global_addr[31:0]
    g0.w = (uint32_t)(ga >> 32) | 0x80000000u;    // addr[56:32] | type=2<<30
    u32x8 g1;
    g1.s0 = 2u << 16;                             // data_size = 4 bytes
    g1.s1 = (uint32_t)PW << 16;                   // tensor_dim0 = 256
    g1.s2 = (uint32_t)PH << 16;                   // tensor_dim1 = 256
    g1.s3 = (uint32_t)TILE << 16;                 // tile_dim0  = 64
    g1.s4 = (uint32_t)TILE;                       // tile_dim1  = 64 (dim2 = 0)
    g1.s5 = (uint32_t)PW;                         // tensor_dim0_stride = 256
    g1.s6 = 0u;
    g1.s7 = 0u;
    asm volatile("tensor_store_from_lds %0, %1" :: "s"(g0), "s"(g1) : "memory");
}

__device__ __forceinline__ float leaky(float v) {
    return fmaxf(v, ALPHA_ * v);   // exact for alpha in (0,1)
}

__global__ __launch_bounds__(NTHR)
void tv_leakyrelu_kernel(const float* __restrict__ z,
                         const float* __restrict__ lam,
                         float* __restrict__ out) {
    extern __shared__ float smem[];
    float* u_s  = smem;
    float* px_s = smem + RR;
    float* py_s = smem + 2*RR;
    float* z_s  = smem + 3*RR;

    const int tid   = threadIdx.x;
    const int plane = blockIdx.x >> 4;      // 512 planes
    const int tile  = blockIdx.x & 15;      // 4x4 tiles of 64x64
    const int x0 = (tile & 3)  * TILE;
    const int y0 = (tile >> 2) * TILE;

    const float lam_c = fminf(fmaxf(lam[0], 0.0f), 1.0f);
    const float k = lam_c * TAU_;

    const float* zplane = z   + (size_t)plane * PLANE;
    float*       oplane = out + (size_t)plane * PLANE;

    // ---- async DMA z region into LDS (clamp-to-edge halo), zero duals -----
    for (int n = tid; n < RR; n += NTHR) {
        int ly = n / R, lx = n - ly * R;
        int gx = min(max(x0 + lx - HALO, 0), PW - 1);
        int gy = min(max(y0 + ly - HALO, 0), PH - 1);
        uint32_t voff = (uint32_t)((gy * PW + gx) * 4);
        async_load_b32(lds_byte_off(&z_s[n]), zplane, voff);
        px_s[n] = 0.0f;
        py_s[n] = 0.0f;
    }
    wait_async0();
    __syncthreads();

    // ---- u = leaky_relu(z) ------------------------------------------------
    for (int n = tid; n < RR; n += NTHR) u_s[n] = leaky(z_s[n]);
    __syncthreads();

    // ---- 10 Chambolle iterations, fully LDS-resident ----------------------
    for (int it = 0; it < NITER; ++it) {
        // pass A: forward-diff gradient, dual ascent, unit-ball projection
        for (int n = tid; n < RR; n += NTHR) {
            int ly = n / R, lx = n - ly * R;
            int gx = x0 + lx - HALO;          // global coords drive Neumann BC
            int gy = y0 + ly - HALO;
            float uc  = u_s[n];
            float uxp = (lx < R-1) ? u_s[n + 1] : uc;   // region-edge clamp:
            float uyp = (ly < R-1) ? u_s[n + R] : uc;   // only invalid halo
            float gxv = (gx == PW-1) ? 0.0f : (uxp - uc);
            float gyv = (gy == PH-1) ? 0.0f : (uyp - uc);
            float npx = px_s[n] + k * gxv;
            float npy = py_s[n] + k * gyv;
            float n2  = npx*npx + npy*npy + EPS_;
            // 1/max(sqrt(n2),1) == min(rsqrt(n2),1)
            float inv = fminf(__builtin_amdgcn_rsqf(n2), 1.0f);
            px_s[n] = npx * inv;
            py_s[n] = npy * inv;
        }
        __syncthreads();
        // pass B: backward-diff divergence, primal leaky-ReLU step
        for (int n = tid; n < RR; n += NTHR) {
            int ly = n / R, lx = n - ly * R;
            int gx = x0 + lx - HALO;
            int gy = y0 + ly - HALO;
            float pxc = px_s[n];
            float pyc = py_s[n];
            float pxl = (lx > 0) ? px_s[n - 1] : pxc;
            float pyl = (ly > 0) ? py_s[n - R] : pyc;
            float dvx = (gx == 0) ? pxc : (pxc - pxl);
            float dvy = (gy == 0) ? pyc : (pyc - pyl);
            u_s[n] = leaky(z_s[n] + lam_c * (dvx + dvy));
        }
        __syncthreads();
    }

    // ---- compact interior 64x64 of u into a contiguous LDS staging area ---
    // (TDM stores read LDS contiguously; padding exists only on loads.)
    // px_s is dead now; reuse it as the staging buffer (4096 <= RR floats).
    float* stage = px_s;
    for (int m = tid; m < TILE*TILE; m += NTHR) {
        int iy = m >> 6, ix = m & 63;
        stage[m] = u_s[(iy + HALO) * R + (ix + HALO)];
    }
    __syncthreads();

    // ---- one Tensor Data Mover store per workgroup ------------------------
    if (tid < 32) {                        // wave 0 only (EXEC ignored by TDM)
        tdm_store_tile64(lds_byte_off(stage), oplane + y0 * PW + x0);
        __builtin_amdgcn_s_wait_tensorcnt(0);
    }
}

extern "C" void kernel_launch(void* const* d_in, const int* in_sizes, int n_in,
                              void* d_out, int out_size, void* d_ws, size_t ws_size,
                              hipStream_t stream) {
    const float* z   = (const float*)d_in[0];   // (8,64,256,256) fp32
    const float* lam = (const float*)d_in[1];   // scalar fp32
    float* out = (float*)d_out;

    // 110 KB dynamic LDS per workgroup (320 KB available per WGP on CDNA5)
    (void)hipFuncSetAttribute((const void*)tv_leakyrelu_kernel,
                              hipFuncAttributeMaxDynamicSharedMemorySize,
                              (int)SMEM_BYTES);

    const int planes = 8 * 64;                  // 512 independent 256x256 planes
    const int blocks = planes * 16;             // 4x4 tiles per plane
    tv_leakyrelu_kernel<<<blocks, NTHR, SMEM_BYTES, stream>>>(z, lam, out);
    (void)in_sizes; (void)n_in; (void)out_size; (void)d_ws; (void)ws_size;
}